// GroupKANLayer_7954279432576
// MI455X (gfx1250) — compile-verified
//
#include <hip/hip_runtime.h>
#include <hip/hip_bf16.h>

typedef float v2f __attribute__((ext_vector_type(2)));
typedef float v8f __attribute__((ext_vector_type(8)));

#define G_     8
#define GIN_   256
#define GOUT_  256
#define NROWS_ 8192
#define NCOLS_ 2048

#define BM 128           // output rows per block
#define BN 64            // output cols per block
#define BK 64            // K chunk staged in LDS
#define LDX 68           // padded LDS stride (floats): 16B aligned rows, conflict-free A frags
#define LDW 68

__global__ __launch_bounds__(256, 2)
void kan_gemm_rational(const float* __restrict__ x,
                       const float* __restrict__ W,
                       const float* __restrict__ b,
                       const float* __restrict__ p,
                       const float* __restrict__ q,
                       float* __restrict__ out)
{
    __shared__ float xs[BM * LDX];   // 128 x 64 x-tile (padded)
    __shared__ float ws[BK * LDW];   // 64 x 64 W-tile (padded)

    const int cb = blockIdx.x;              // 0..31 column block (64 cols each)
    const int rb = blockIdx.y;              // 0..63 row block
    const int g  = cb >> 2;                 // group id (4 col blocks per group)
    const int nbase_in_g = (cb & 3) * BN;   // col base inside group
    const int row0 = rb * BM;

    const int tid   = threadIdx.x;
    const int lane  = tid & 31;
    const int wave  = tid >> 5;             // 0..7
    const int wm    = wave >> 2;            // 0..1  -> 64-row half of the tile
    const int wn    = wave & 3;             // 0..3  -> 16-col slice
    const int lhalf = lane >> 4;            // 0/1   -> K half for A/B frags, M half for C/D
    const int l16   = lane & 15;

    v8f acc0 = {}, acc1 = {}, acc2 = {}, acc3 = {};

    const float* xg = x + (size_t)row0 * NCOLS_ + g * GIN_;
    const float* Wg = W + (size_t)g * GIN_ * GOUT_ + nbase_in_g;

    const int r0 = tid >> 4;          // 0..15 loader row
    const int c4 = (tid & 15) * 4;    // 0..60 loader col (float4)

    for (int kb = 0; kb < GIN_; kb += BK) {
        // ---- stage x tile: 128 x 64, 8 float4 per thread ----
        #pragma unroll
        for (int pass = 0; pass < 8; ++pass) {
            const int r = pass * 16 + r0;
            const float4 v = *(const float4*)(xg + (size_t)r * NCOLS_ + kb + c4);
            *(float4*)&xs[r * LDX + c4] = v;
        }
        // ---- stage W tile: 64 x 64, 4 float4 per thread ----
        #pragma unroll
        for (int pass = 0; pass < 4; ++pass) {
            const int r = pass * 16 + r0;
            const float4 v = *(const float4*)(Wg + (size_t)(kb + r) * GOUT_ + c4);
            *(float4*)&ws[r * LDW + c4] = v;
        }
        // prefetch next K chunk of x into cache while we compute this one
        if (kb + BK < GIN_) {
            __builtin_prefetch(xg + (size_t)r0 * NCOLS_ + kb + BK + c4, 0, 3);
        }
        __syncthreads();

        // ---- 16 K-steps of 4, 4 stacked M subtiles per wave ----
        #pragma unroll 4
        for (int k = 0; k < BK; k += 4) {
            const int ka = k + 2 * lhalf;  // this lane's K offset within the 4-wide step

            v2f bf;                         // B frag: rows ka, ka+1 of W tile, col = wn*16+l16
            bf.x = ws[(ka + 0) * LDW + wn * 16 + l16];
            bf.y = ws[(ka + 1) * LDW + wn * 16 + l16];

            const int mb = wm * 64 + l16;   // A frag rows for the 4 subtiles
            v2f a0 = *(const v2f*)&xs[(mb +  0) * LDX + ka];
            v2f a1 = *(const v2f*)&xs[(mb + 16) * LDX + ka];
            v2f a2 = *(const v2f*)&xs[(mb + 32) * LDX + ka];
            v2f a3 = *(const v2f*)&xs[(mb + 48) * LDX + ka];

            acc0 = __builtin_amdgcn_wmma_f32_16x16x4_f32(false, a0, false, bf, (short)0, acc0, false, false);
            acc1 = __builtin_amdgcn_wmma_f32_16x16x4_f32(false, a1, false, bf, (short)0, acc1, false, false);
            acc2 = __builtin_amdgcn_wmma_f32_16x16x4_f32(false, a2, false, bf, (short)0, acc2, false, false);
            acc3 = __builtin_amdgcn_wmma_f32_16x16x4_f32(false, a3, false, bf, (short)0, acc3, false, false);
        }
        __syncthreads();
    }

    // ---- epilogue: bias + rational, direct from C/D layout ----
    const int gout_col = nbase_in_g + wn * 16 + l16;      // col within group
    const float bias = b[g * GOUT_ + gout_col];
    const float p0 = p[g * 4 + 0], p1 = p[g * 4 + 1];
    const float p2 = p[g * 4 + 2], p3 = p[g * 4 + 3];
    const float q0 = q[g * 3 + 0], q1 = q[g * 3 + 1], q2 = q[g * 3 + 2];

    float* outp = out + (size_t)row0 * NCOLS_ + g * GOUT_ + gout_col;

    const v8f accs[4] = {acc0, acc1, acc2, acc3};
    #pragma unroll
    for (int i = 0; i < 4; ++i) {
        #pragma unroll
        for (int j = 0; j < 8; ++j) {
            const int m = wm * 64 + i * 16 + 8 * lhalf + j;  // output row within tile
            const float y  = accs[i][j] + bias;
            const float y2 = y * y;
            const float y3 = y2 * y;
            const float num = p0 + p1 * y + p2 * y2 + p3 * y3;
            const float den = 1.0f + fabsf(q0 * y + q1 * y2 + q2 * y3);
            outp[(size_t)m * NCOLS_] = num / den;
        }
    }
}

extern "C" void kernel_launch(void* const* d_in, const int* in_sizes, int n_in,
                              void* d_out, int out_size, void* d_ws, size_t ws_size,
                              hipStream_t stream) {
    const float* x = (const float*)d_in[0];
    const float* W = (const float*)d_in[1];
    const float* b = (const float*)d_in[2];
    const float* p = (const float*)d_in[3];
    const float* q = (const float*)d_in[4];
    float* out = (float*)d_out;

    dim3 grid(NCOLS_ / BN, NROWS_ / BM);   // (32, 64)
    dim3 block(256);
    kan_gemm_rational<<<grid, block, 0, stream>>>(x, W, b, p, q, out);
}